// CpGPredictor_34677565948729
// MI455X (gfx1250) — compile-verified
//
#include <hip/hip_runtime.h>
#include <hip/hip_bf16.h>

typedef __attribute__((ext_vector_type(2))) float v2f;
typedef __attribute__((ext_vector_type(8))) float v8f;

#define EMBED 128
#define VOCAB 5
#define SEQ   8192
#define BATCH 128

// ---------------------------------------------------------------------------
// Kernel 1: one wave32 computes the 10-entry LUT with V_WMMA_F32_16X16X4_F32.
//   D = A x B accumulated over K=128 in steps of 4.
//   A (16x4 f32): row 0 = w_prev, row 1 = w_curr, rows 2..15 = 0.
//   B (4x16 f32): col v = emb_table[v] (v<5), cols 5..15 = 0.
//   D[0][v] = p[v], D[1][v] = c[v].
// VGPR layouts per CDNA5 ISA 7.12.2:
//   A: lane L(0..15) holds A[M=L][k0+0], A[M=L][k0+1]; lanes 16..31 hold K=k0+2,k0+3
//   B: lane L(0..15) holds B[k0+0][N=L], B[k0+1][N=L]; lanes 16..31 hold K=k0+2,k0+3
//   D: lane L(0..15) vgpr r -> D[M=r][N=L]
// ---------------------------------------------------------------------------
__global__ __launch_bounds__(32) void cpg_lut_wmma(
    const float* __restrict__ emb,    // [VOCAB][EMBED]
    const float* __restrict__ lin_w,  // [2*EMBED] : w_prev | w_curr
    const float* __restrict__ lin_b,  // [1]
    float*       __restrict__ lut)    // out: [10] = p[0..4], c'[0..4]
{
    const int lane = threadIdx.x;      // 0..31
    const int m    = lane & 15;        // A-row / B-col index for this lane
    const int half = lane >> 4;        // 0: K=k0+0,k0+1   1: K=k0+2,k0+3

    v8f acc = {};
    for (int k0 = 0; k0 < EMBED; k0 += 4) {
        const int ka = k0 + (half << 1);

        float a0 = 0.0f, a1 = 0.0f;
        if (m < 2) {                       // only rows 0 (w_prev) and 1 (w_curr)
            const float* wrow = lin_w + m * EMBED;
            a0 = wrow[ka];
            a1 = wrow[ka + 1];
        }
        float b0 = 0.0f, b1 = 0.0f;
        if (m < VOCAB) {                   // only 5 valid embedding columns
            const float* erow = emb + m * EMBED;
            b0 = erow[ka];
            b1 = erow[ka + 1];
        }

        v2f a = {a0, a1};
        v2f b = {b0, b1};
        // 8 args: (neg_a, A, neg_b, B, c_mod, C, reuse_a, reuse_b)
        acc = __builtin_amdgcn_wmma_f32_16x16x4_f32(
            false, a, false, b, (short)0, acc, false, false);
    }

    if (lane < VOCAB) {
        lut[lane]         = acc[0];               // p[v]   = D[0][v]
        lut[VOCAB + lane] = acc[1] + lin_b[0];    // c'[v]  = D[1][v] + bias
    }
}

// ---------------------------------------------------------------------------
// Kernel 2: streaming pass, out[b][s] = p[x[b][s]] + c'[x[b][s+1]].
// Bandwidth-bound: ~4 MB reads + ~4.2 MB writes -> ~0.35 us at 23.3 TB/s.
// One output per thread; fully coalesced b32 loads/stores; LUT in LDS
// (10 distinct banks, same-address lane reads broadcast).
// ---------------------------------------------------------------------------
__global__ __launch_bounds__(256) void cpg_logits(
    const int*   __restrict__ x,    // [BATCH][SEQ] int32
    const float* __restrict__ lut,  // [10]
    float*       __restrict__ out)  // [BATCH][SEQ-1]
{
    __shared__ float l[2 * VOCAB];
    if (threadIdx.x < 2 * VOCAB) l[threadIdx.x] = lut[threadIdx.x];
    __syncthreads();

    const int s = blockIdx.x * blockDim.x + threadIdx.x;
    const int b = blockIdx.y;
    if (s < SEQ - 1) {
        const int* row = x + (size_t)b * SEQ;
        const int v0 = row[s];
        const int v1 = row[s + 1];
        out[(size_t)b * (SEQ - 1) + s] = l[v0] + l[VOCAB + v1];
    }
}

extern "C" void kernel_launch(void* const* d_in, const int* in_sizes, int n_in,
                              void* d_out, int out_size, void* d_ws, size_t ws_size,
                              hipStream_t stream) {
    const int*   x     = (const int*)  d_in[0];   // [128][8192] int
    const float* emb   = (const float*)d_in[1];   // [5][128]
    const float* lin_w = (const float*)d_in[2];   // [256]
    const float* lin_b = (const float*)d_in[3];   // [1]
    float*       out   = (float*)d_out;           // [128][8191]
    float*       lut   = (float*)d_ws;            // 10 floats of scratch

    (void)in_sizes; (void)n_in; (void)out_size; (void)ws_size;

    cpg_lut_wmma<<<dim3(1), dim3(32), 0, stream>>>(emb, lin_w, lin_b, lut);

    dim3 grid((SEQ - 1 + 255) / 256, BATCH);   // (32, 128)
    cpg_logits<<<grid, dim3(256), 0, stream>>>(x, lut, out);
}